// Instance_Aware_Backbone_47863115546698
// MI455X (gfx1250) — compile-verified
//
#include <hip/hip_runtime.h>

#define NPT   65536
#define KTAP  27
#define CIN   4
#define CCH   32
#define NWIN  2048
#define CAPW  32
#define NHEAD 2
#define HDIM  16
#define RSTR  36  // padded LDS row stride: 16B-aligned stores, conflict-free A reads

typedef __attribute__((ext_vector_type(2))) float v2f;
typedef __attribute__((ext_vector_type(8))) float v8f;

// f32 WMMA: D(16x16) = A(16x4) * B(4x16) + C
static __device__ __forceinline__ v8f wmma4(v2f a, v2f b, v8f c) {
  return __builtin_amdgcn_wmma_f32_16x16x4_f32(false, a, false, b, (short)0, c,
                                               false, false);
}

// ---------------- input conv: feats[N,4] -> x[N,32] --------------------------
__global__ __launch_bounds__(256) void k_conv_in(const float* __restrict__ feats,
                                                 const int* __restrict__ pairs,
                                                 const float* __restrict__ w_in,
                                                 float* __restrict__ out) {
  __shared__ float wl[KTAP * CIN * CCH];  // 3456 floats
  int tid = threadIdx.x;
  for (int i = tid; i < KTAP * CIN * CCH; i += 256) wl[i] = w_in[i];
  __syncthreads();

  int wv = tid >> 5, lane = tid & 31, half = lane >> 4, m = lane & 15;
  int n0 = (blockIdx.x * 8 + wv) * 16;
  int c0 = half * 2;

  v8f acc0 = {0, 0, 0, 0, 0, 0, 0, 0};
  v8f acc1 = {0, 0, 0, 0, 0, 0, 0, 0};
  for (int k = 0; k < KTAP; ++k) {
    int pr = pairs[k * NPT + n0 + m];
    float msk = (pr >= 0) ? 1.f : 0.f;
    const float* fr = feats + (pr >= 0 ? pr : 0) * CIN;
    v2f a = {fr[c0] * msk, fr[c0 + 1] * msk};
    const float* wk = wl + k * CIN * CCH;
    v2f b0 = {wk[c0 * CCH + m], wk[(c0 + 1) * CCH + m]};
    v2f b1 = {wk[c0 * CCH + 16 + m], wk[(c0 + 1) * CCH + 16 + m]};
    acc0 = wmma4(a, b0, acc0);
    acc1 = wmma4(a, b1, acc1);
  }
#pragma unroll
  for (int r = 0; r < 8; ++r) {
    int row = n0 + r + half * 8;
    out[row * CCH + m] = acc0[r];
    out[row * CCH + 16 + m] = acc1[r];
  }
}

// ------- fused bnrelu(x)->gather->32x32 conv, optional residual add ----------
__global__ __launch_bounds__(256) void k_conv32(const float* __restrict__ xin,
                                                const int* __restrict__ pairs,
                                                const float* __restrict__ g,
                                                const float* __restrict__ bb,
                                                const float* __restrict__ w,
                                                const float* __restrict__ resid,
                                                float* __restrict__ out) {
  __shared__ float wl[9 * CCH * CCH];    // 36 KB: 9 weight taps staged
  __shared__ float rows[8 * 16 * RSTR];  // 18 KB: per-wave activated row tile
  int tid = threadIdx.x, wv = tid >> 5, lane = tid & 31;
  int half = lane >> 4, m = lane & 15;
  int n0 = (blockIdx.x * 8 + wv) * 16;
  float* rw = rows + wv * 16 * RSTR;

  int cld = (lane & 7) * 4;  // staging column (per-lane constant)
  int rld = lane >> 3;       // staging row within group of 4
  float4 gv = *(const float4*)(g + cld);
  float4 bv = *(const float4*)(bb + cld);

  v8f acc0 = {0, 0, 0, 0, 0, 0, 0, 0};
  v8f acc1 = {0, 0, 0, 0, 0, 0, 0, 0};

  for (int stage = 0; stage < 3; ++stage) {
    __syncthreads();
    const float* wsrc = w + stage * 9 * CCH * CCH;
    for (int i = tid; i < 9 * CCH * CCH; i += 256) wl[i] = wsrc[i];
    __syncthreads();

    for (int kt = 0; kt < 9; ++kt) {
      int k = stage * 9 + kt;
      const int* pk = pairs + k * NPT + n0;
      // coalesced gather of 16 rows (1 cacheline req per row) + fused BN/ReLU
#pragma unroll
      for (int it = 0; it < 4; ++it) {
        int r = it * 4 + rld;
        int pr = pk[r];
        float msk = (pr >= 0) ? 1.f : 0.f;
        float4 v = *(const float4*)(xin + (pr >= 0 ? pr : 0) * CCH + cld);
        float4 s;
        s.x = fmaxf(fmaf(v.x, gv.x, bv.x), 0.f) * msk;
        s.y = fmaxf(fmaf(v.y, gv.y, bv.y), 0.f) * msk;
        s.z = fmaxf(fmaf(v.z, gv.z, bv.z), 0.f) * msk;
        s.w = fmaxf(fmaf(v.w, gv.w, bv.w), 0.f) * msk;
        *(float4*)(rw + r * RSTR + cld) = s;
      }
      const float* wk = wl + kt * CCH * CCH;
#pragma unroll
      for (int kk = 0; kk < 8; ++kk) {
        int c0 = kk * 4 + half * 2;
        v2f a = {rw[m * RSTR + c0], rw[m * RSTR + c0 + 1]};
        v2f b0 = {wk[c0 * CCH + m], wk[(c0 + 1) * CCH + m]};
        v2f b1 = {wk[c0 * CCH + 16 + m], wk[(c0 + 1) * CCH + 16 + m]};
        acc0 = wmma4(a, b0, acc0);
        acc1 = wmma4(a, b1, acc1);
      }
    }
  }
#pragma unroll
  for (int r = 0; r < 8; ++r) {
    int row = n0 + r + half * 8;
    int i0 = row * CCH + m, i1 = row * CCH + 16 + m;
    float r0 = acc0[r], r1 = acc1[r];
    if (resid) {
      r0 += resid[i0];
      r1 += resid[i1];
    }
    out[i0] = r0;
    out[i1] = r1;
  }
}

// ---------------- windowed sphere attention (one wave per window) ------------
__global__ __launch_bounds__(64) void k_attn(
    const float* __restrict__ x, const int* __restrict__ win_idx,
    const int* __restrict__ rel_idx, const float* __restrict__ w_qkv,
    const float* __restrict__ b_qkv, const float* __restrict__ w_proj,
    const float* __restrict__ b_proj, const float* __restrict__ tq,
    const float* __restrict__ tk, const float* __restrict__ tv,
    float* __restrict__ out) {
  __shared__ float smem[2 * 6144];  // 48 KB: 2 windows per block
  int wib = threadIdx.x >> 5, lane = threadIdx.x & 31;
  int half = lane >> 4, m = lane & 15;
  int w = blockIdx.x * 2 + wib;

  float* xw = smem + wib * 6144;  // [32 tok][32 ch]
  float* qkv = xw + 1024;         // [32 tok][96]
  float* att = qkv + 3072;        // [32 i][32 j] (per-head, reused)
  float* ob = att + 1024;         // [32 tok][32 ch]

  {
    const float* xr = x + win_idx[w * CAPW + lane] * CCH;
#pragma unroll
    for (int c = 0; c < 8; ++c)
      *(float4*)(xw + lane * CCH + c * 4) = *(const float4*)(xr + c * 4);
  }

  // QKV projection (M=32, N=96, K=32), q pre-scaled by HD^-0.5
#pragma unroll
  for (int mt = 0; mt < 2; ++mt) {
#pragma unroll
    for (int nh = 0; nh < 6; ++nh) {
      int col = nh * 16 + m;
      float bias = b_qkv[col];
      v8f acc = {bias, bias, bias, bias, bias, bias, bias, bias};
#pragma unroll
      for (int kk = 0; kk < 8; ++kk) {
        int c0 = kk * 4 + half * 2;
        v2f a = {xw[(mt * 16 + m) * CCH + c0], xw[(mt * 16 + m) * CCH + c0 + 1]};
        v2f b = {w_qkv[c0 * 96 + col], w_qkv[(c0 + 1) * 96 + col]};
        acc = wmma4(a, b, acc);
      }
      float scale = (nh < 2) ? 0.25f : 1.f;
#pragma unroll
      for (int r = 0; r < 8; ++r)
        qkv[(mt * 16 + r + half * 8) * 96 + col] = acc[r] * scale;
    }
  }

  const int* relw = rel_idx + w * CAPW * CAPW;

  for (int h = 0; h < NHEAD; ++h) {
    int hb = h * HDIM;
    // scores = q @ k^T  (M=32, N=32, K=16)
#pragma unroll
    for (int it = 0; it < 2; ++it) {
#pragma unroll
      for (int jt = 0; jt < 2; ++jt) {
        v8f acc = {0, 0, 0, 0, 0, 0, 0, 0};
#pragma unroll
        for (int kk = 0; kk < 4; ++kk) {
          int d0 = kk * 4 + half * 2;
          v2f a = {qkv[(it * 16 + m) * 96 + hb + d0],
                   qkv[(it * 16 + m) * 96 + hb + d0 + 1]};
          v2f b = {qkv[(jt * 16 + m) * 96 + 32 + hb + d0],
                   qkv[(jt * 16 + m) * 96 + 32 + hb + d0 + 1]};
          acc = wmma4(a, b, acc);
        }
#pragma unroll
        for (int r = 0; r < 8; ++r)
          att[(it * 16 + r + half * 8) * CAPW + jt * 16 + m] = acc[r];
      }
    }
    // relative-position score terms + fused softmax (lane = column j)
    float kreg[HDIM];
#pragma unroll
    for (int d = 0; d < HDIM; ++d) kreg[d] = qkv[lane * 96 + 32 + hb + d];
    for (int i = 0; i < CAPW; ++i) {
      int rel = relw[i * CAPW + lane];
      const float* tqp = tq + rel * NHEAD * HDIM + hb;
      const float* tkp = tk + rel * NHEAD * HDIM + hb;
      float s = att[i * CAPW + lane];
#pragma unroll
      for (int d = 0; d < HDIM; ++d)
        s += qkv[i * 96 + hb + d] * tqp[d] + kreg[d] * tkp[d];
      float mx = s;
#pragma unroll
      for (int off = 16; off > 0; off >>= 1)
        mx = fmaxf(mx, __shfl_xor(mx, off, 32));
      float e = __expf(s - mx);
      float sum = e;
#pragma unroll
      for (int off = 16; off > 0; off >>= 1) sum += __shfl_xor(sum, off, 32);
      att[i * CAPW + lane] = e / sum;
    }
    // out = attn @ v  (M=32, N=16, K=32)
#pragma unroll
    for (int it = 0; it < 2; ++it) {
      v8f acc = {0, 0, 0, 0, 0, 0, 0, 0};
#pragma unroll
      for (int kk = 0; kk < 8; ++kk) {
        int j0 = kk * 4 + half * 2;
        v2f a = {att[(it * 16 + m) * CAPW + j0],
                 att[(it * 16 + m) * CAPW + j0 + 1]};
        v2f b = {qkv[j0 * 96 + 64 + hb + m], qkv[(j0 + 1) * 96 + 64 + hb + m]};
        acc = wmma4(a, b, acc);
      }
#pragma unroll
      for (int r = 0; r < 8; ++r)
        ob[(it * 16 + r + half * 8) * CCH + hb + m] = acc[r];
    }
    // + attn-weighted tv gather (lane = half*16 + d)
    for (int i = 0; i < CAPW; ++i) {
      float part = 0.f;
#pragma unroll
      for (int jj = 0; jj < 16; ++jj) {
        int j = half * 16 + jj;
        int rel = relw[i * CAPW + j];
        part += att[i * CAPW + j] * tv[rel * NHEAD * HDIM + hb + m];
      }
      part += __shfl_xor(part, 16, 32);
      if (half == 0) ob[i * CCH + hb + m] += part;
    }
  }

  // output projection + residual; win_idx is a permutation -> exclusive scatter
#pragma unroll
  for (int mt = 0; mt < 2; ++mt) {
#pragma unroll
    for (int nh = 0; nh < 2; ++nh) {
      int col = nh * 16 + m;
      float bias = b_proj[col];
      v8f acc = {bias, bias, bias, bias, bias, bias, bias, bias};
#pragma unroll
      for (int kk = 0; kk < 8; ++kk) {
        int c0 = kk * 4 + half * 2;
        v2f a = {ob[(mt * 16 + m) * CCH + c0], ob[(mt * 16 + m) * CCH + c0 + 1]};
        v2f b = {w_proj[c0 * CCH + col], w_proj[(c0 + 1) * CCH + col]};
        acc = wmma4(a, b, acc);
      }
#pragma unroll
      for (int r = 0; r < 8; ++r) {
        int tok = mt * 16 + r + half * 8;
        int row = win_idx[w * CAPW + tok];
        out[row * CCH + col] = x[row * CCH + col] + acc[r];
      }
    }
  }
}

// ---------------------------------------------------------------------------
extern "C" void kernel_launch(void* const* d_in, const int* in_sizes, int n_in,
                              void* d_out, int out_size, void* d_ws,
                              size_t ws_size, hipStream_t stream) {
  (void)in_sizes; (void)n_in; (void)out_size; (void)ws_size;
  const float* feats   = (const float*)d_in[0];
  const int*   pairs   = (const int*)d_in[1];
  const int*   win_idx = (const int*)d_in[2];
  const int*   rel_idx = (const int*)d_in[3];
  const float* w_in    = (const float*)d_in[4];
  const float* g11 = (const float*)d_in[5];
  const float* b11 = (const float*)d_in[6];
  const float* w11 = (const float*)d_in[7];
  const float* g12 = (const float*)d_in[8];
  const float* b12 = (const float*)d_in[9];
  const float* w12 = (const float*)d_in[10];
  const float* g21 = (const float*)d_in[11];
  const float* b21 = (const float*)d_in[12];
  const float* w21 = (const float*)d_in[13];
  const float* g22 = (const float*)d_in[14];
  const float* b22 = (const float*)d_in[15];
  const float* w22 = (const float*)d_in[16];
  const float* w_qkv  = (const float*)d_in[17];
  const float* b_qkv  = (const float*)d_in[18];
  const float* w_proj = (const float*)d_in[19];
  const float* b_proj = (const float*)d_in[20];
  const float* tq = (const float*)d_in[21];
  const float* tk = (const float*)d_in[22];
  const float* tv = (const float*)d_in[23];

  float* xb = (float*)d_ws;    // N x 32
  float* tb = xb + NPT * CCH;  // N x 32
  float* out = (float*)d_out;

  const int convGrid = NPT / 128;  // 8 waves/block, 16 pts/wave

  k_conv_in<<<convGrid, 256, 0, stream>>>(feats, pairs, w_in, xb);
  // residual block 1 (BN+ReLU fused into the conv gather stage)
  k_conv32<<<convGrid, 256, 0, stream>>>(xb, pairs, g11, b11, w11, nullptr, tb);
  k_conv32<<<convGrid, 256, 0, stream>>>(tb, pairs, g12, b12, w12, xb, xb);
  // residual block 2
  k_conv32<<<convGrid, 256, 0, stream>>>(xb, pairs, g21, b21, w21, nullptr, tb);
  k_conv32<<<convGrid, 256, 0, stream>>>(tb, pairs, g22, b22, w22, xb, xb);
  // windowed attention + residual -> final output
  k_attn<<<NWIN / 2, 64, 0, stream>>>(xb, win_idx, rel_idx, w_qkv, b_qkv,
                                      w_proj, b_proj, tq, tk, tv, out);
}